// GraphEncoder_25993142075733
// MI455X (gfx1250) — compile-verified
//
#include <hip/hip_runtime.h>
#include <hip/hip_bf16.h>

typedef float v2f __attribute__((ext_vector_type(2)));
typedef float v8f __attribute__((ext_vector_type(8)));

#define BN_SCALE 0.99999500003749937f

// ---------------------------------------------------------------------------
// Edge message kernel: for each edge e, ea = eb + edge_attr[e] @ ew  (16 x d),
// msg = relu(h[src] + ea), atomically scatter-add msg into agg[dst].
// threads_per_edge = d/4; each thread owns 4 consecutive channels.
// ---------------------------------------------------------------------------
__global__ __launch_bounds__(256) void edge_msg_kernel(
    const float* __restrict__ h, int d,
    const float* __restrict__ edge_attr,
    const int* __restrict__ src, const int* __restrict__ dst,
    const float* __restrict__ ew, const float* __restrict__ eb,
    float* __restrict__ agg, int E)
{
    __shared__ float sw[16 * 128];
    __shared__ float sb[128];
    int t = threadIdx.x;
    for (int i = t; i < 16 * d; i += blockDim.x) sw[i] = ew[i];
    for (int i = t; i < d; i += blockDim.x) sb[i] = eb[i];
    __syncthreads();

    int tpe = d >> 2;                       // threads per edge (16 or 32)
    long long gid = (long long)blockIdx.x * blockDim.x + t;
    long long e = gid / tpe;
    int cg = (int)(gid % tpe);
    if (e >= E) return;
    int j0 = cg * 4;

    const float4* ap4 = (const float4*)(edge_attr + e * 16);  // one 64B line
    float4 aq[4];
    #pragma unroll
    for (int q = 0; q < 4; ++q) aq[q] = ap4[q];
    float a[16] = { aq[0].x, aq[0].y, aq[0].z, aq[0].w,
                    aq[1].x, aq[1].y, aq[1].z, aq[1].w,
                    aq[2].x, aq[2].y, aq[2].z, aq[2].w,
                    aq[3].x, aq[3].y, aq[3].z, aq[3].w };

    float acc0 = sb[j0], acc1 = sb[j0 + 1], acc2 = sb[j0 + 2], acc3 = sb[j0 + 3];
    #pragma unroll
    for (int k = 0; k < 16; ++k) {
        float av = a[k];
        const float* wr = &sw[k * d + j0];
        acc0 = fmaf(av, wr[0], acc0);
        acc1 = fmaf(av, wr[1], acc1);
        acc2 = fmaf(av, wr[2], acc2);
        acc3 = fmaf(av, wr[3], acc3);
    }

    int s = src[e];
    int dn = dst[e];
    const float4 hv = *(const float4*)(h + (size_t)s * d + j0);
    float m0 = fmaxf(hv.x + acc0, 0.0f);
    float m1 = fmaxf(hv.y + acc1, 0.0f);
    float m2 = fmaxf(hv.z + acc2, 0.0f);
    float m3 = fmaxf(hv.w + acc3, 0.0f);
    float* ag = agg + (size_t)dn * d + j0;
    atomicAdd(ag + 0, m0);
    atomicAdd(ag + 1, m1);
    atomicAdd(ag + 2, m2);
    atomicAdd(ag + 3, m3);
}

// ---------------------------------------------------------------------------
// WMMA fp32 GEMM: out[nrows,128] = epilogue( (A0 (+A1)) [nrows,d] @ W [d,128] )
// epilogue: v = acc + bias; if gamma: v = v*(gamma*BN_SCALE)+beta; if relu: max(v,0)
// 8 waves/block; each wave computes a 16-row strip x all 128 columns with
// V_WMMA_F32_16X16X4_F32. W is staged in LDS in fragment-ready float2 pairs
// so each B fragment is one ds_load_b64. Branch-free lane predication
// (row clamp) keeps EXEC all-ones through the WMMA loop.
// out may alias A0 or A1 (each wave reads its rows fully before writing them).
// ---------------------------------------------------------------------------
__global__ __launch_bounds__(256) void gemm_node_kernel(
    const float* A0, const float* A1,
    const float* __restrict__ W, const float* __restrict__ bias,
    const float* __restrict__ gamma, const float* __restrict__ beta,
    float* out, int nrows, int d, int do_relu)
{
    // sw[k2*128 + n] = { W[2*k2][n], W[2*k2+1][n] }  (64 KB for d=128)
    __shared__ v2f sw[64 * 128];
    int t = threadIdx.x;
    int npairs = (d >> 1) * 128;
    for (int i = t; i < npairs; i += 256) {
        int k2 = i >> 7, n = i & 127;
        sw[i] = (v2f){ W[(2 * k2) * 128 + n], W[(2 * k2 + 1) * 128 + n] };
    }
    __syncthreads();

    int wave = t >> 5;                      // wave32
    int lane = t & 31;
    int row0 = (blockIdx.x * 8 + wave) * 16;
    if (row0 >= nrows) return;              // wave-uniform exit: EXEC stays full

    int l15   = lane & 15;
    int khalf = lane >> 4;                  // 0 or 1 -> which k-pair of the block
    int arow  = row0 + l15;
    if (arow >= nrows) arow = nrows - 1;    // branch-free clamp (nrows%16==0 here)

    v8f acc[8];
    #pragma unroll
    for (int i = 0; i < 8; ++i) acc[i] = (v8f){0,0,0,0,0,0,0,0};

    const float* a0p = A0 + (size_t)arow * d + 2 * khalf;

    if (A1) {
        const float* a1p = A1 + (size_t)arow * d + 2 * khalf;
        float2 xa = *(const float2*)(a0p);
        float2 xb = *(const float2*)(a1p);
        for (int k0 = 0; k0 < d; k0 += 4) {
            v2f afrag = (v2f){ xa.x + xb.x, xa.y + xb.y };
            if (k0 + 4 < d) {               // uniform: prefetch next A fragment
                xa = *(const float2*)(a0p + k0 + 4);
                xb = *(const float2*)(a1p + k0 + 4);
            }
            const v2f* bp = sw + ((k0 >> 1) + khalf) * 128 + l15;
            #pragma unroll
            for (int nt = 0; nt < 8; ++nt)
                acc[nt] = __builtin_amdgcn_wmma_f32_16x16x4_f32(
                    false, afrag, false, bp[nt * 16], (short)0, acc[nt], false, false);
        }
    } else {
        float2 xa = *(const float2*)(a0p);
        for (int k0 = 0; k0 < d; k0 += 4) {
            v2f afrag = (v2f){ xa.x, xa.y };
            if (k0 + 4 < d)
                xa = *(const float2*)(a0p + k0 + 4);
            const v2f* bp = sw + ((k0 >> 1) + khalf) * 128 + l15;
            #pragma unroll
            for (int nt = 0; nt < 8; ++nt)
                acc[nt] = __builtin_amdgcn_wmma_f32_16x16x4_f32(
                    false, afrag, false, bp[nt * 16], (short)0, acc[nt], false, false);
        }
    }

    // D layout: VGPR r, lanes 0-15 -> M=r, lanes 16-31 -> M=r+8; N = lane&15
    int mbase = row0 + khalf * 8;
    #pragma unroll
    for (int r = 0; r < 8; ++r) {
        int row = mbase + r;
        if (row < nrows) {
            float* op = out + (size_t)row * 128;
            #pragma unroll
            for (int nt = 0; nt < 8; ++nt) {
                int n = nt * 16 + l15;
                float v = acc[nt][r] + bias[n];
                if (gamma) v = fmaf(v, gamma[n] * BN_SCALE, beta[n]);
                if (do_relu) v = fmaxf(v, 0.0f);
                op[n] = v;
            }
        }
    }
}

// ---------------------------------------------------------------------------
// Mean-pool: atomic segment-sum of h[N,128] by batch + counts (32 lanes/node).
// ---------------------------------------------------------------------------
__global__ __launch_bounds__(256) void pool_kernel(
    const float* __restrict__ h, const int* __restrict__ batch,
    float* __restrict__ sums, float* __restrict__ cnt, int n)
{
    long long gid = (long long)blockIdx.x * blockDim.x + threadIdx.x;
    long long node = gid >> 5;
    int lane = (int)(gid & 31);
    if (node >= n) return;
    int g = batch[node];
    const float4 v = *(const float4*)(h + (size_t)node * 128 + lane * 4);
    float* s = sums + (size_t)g * 128 + lane * 4;
    atomicAdd(s + 0, v.x);
    atomicAdd(s + 1, v.y);
    atomicAdd(s + 2, v.z);
    atomicAdd(s + 3, v.w);
    if (lane == 0) atomicAdd(cnt + g, 1.0f);
}

__global__ __launch_bounds__(256) void pool_div_kernel(
    float* __restrict__ sums, const float* __restrict__ cnt)
{
    int i = blockIdx.x * blockDim.x + threadIdx.x;
    if (i < 256 * 128) sums[i] /= fmaxf(cnt[i >> 7], 1.0f);
}

// ---------------------------------------------------------------------------
extern "C" void kernel_launch(void* const* d_in, const int* in_sizes, int n_in,
                              void* d_out, int out_size, void* d_ws, size_t ws_size,
                              hipStream_t stream)
{
    const int N = in_sizes[0] / 64;          // 100000
    const int E = in_sizes[1] / 16;          // 1600000

    const float* x         = (const float*)d_in[0];
    const float* edge_attr = (const float*)d_in[1];
    const int*   eidx      = (const int*)d_in[2];
    const int*   batch     = (const int*)d_in[3];
    const int* src = eidx;
    const int* dst = eidx + E;

    // per-layer params: d_in[4+8i .. 11+8i] = ew, eb, w1, b1, g, bt, w2, b2
    const float* P[24];
    for (int i = 0; i < 24; ++i) P[i] = (const float*)d_in[4 + i];
    const float* o_w1 = (const float*)d_in[28];
    const float* o_b1 = (const float*)d_in[29];
    const float* o_w2 = (const float*)d_in[30];
    const float* o_b2 = (const float*)d_in[31];

    float* bufA   = (float*)d_ws;                        // N*128
    float* bufB   = bufA + (size_t)N * 128;              // N*128
    float* pooled = bufB + (size_t)N * 128;              // 256*128
    float* cnt    = pooled + 256 * 128;                  // 256
    float* tbuf   = cnt + 256;                           // 256*128

    const int gemmGrid = (N + 127) / 128;
    const int in_dims[3] = {64, 128, 128};
    // (h, agg, hh/out) buffer choreography; aliasing is row-safe in gemm kernel
    const float* hcur[3] = {x, bufB, bufA};
    float*       aggs[3] = {bufA, bufA, bufB};
    float*       hout[3] = {bufB, bufA, bufB};

    for (int l = 0; l < 3; ++l) {
        int d = in_dims[l];
        const float* ew = P[8*l+0]; const float* eb = P[8*l+1];
        const float* w1 = P[8*l+2]; const float* b1 = P[8*l+3];
        const float* g  = P[8*l+4]; const float* bt = P[8*l+5];
        const float* w2 = P[8*l+6]; const float* b2 = P[8*l+7];

        hipMemsetAsync(aggs[l], 0, (size_t)N * d * sizeof(float), stream);

        long long ethreads = (long long)E * (d >> 2);
        int eblocks = (int)((ethreads + 255) / 256);
        edge_msg_kernel<<<eblocks, 256, 0, stream>>>(
            hcur[l], d, edge_attr, src, dst, ew, eb, aggs[l], E);

        // (h + agg) @ w1 + b1 -> BN(eval) -> relu   (out may alias agg when d==128)
        gemm_node_kernel<<<gemmGrid, 256, 0, stream>>>(
            hcur[l], aggs[l], w1, b1, g, bt, hout[l], N, d, 1);
        // hh @ w2 + b2 -> relu  (in place)
        gemm_node_kernel<<<gemmGrid, 256, 0, stream>>>(
            hout[l], nullptr, w2, b2, nullptr, nullptr, hout[l], N, 128, 1);
    }

    // mean pooling over batch (h3 lives in bufB)
    hipMemsetAsync(pooled, 0, (256 * 128 + 256) * sizeof(float), stream);
    {
        long long pthreads = (long long)N * 32;
        int pblocks = (int)((pthreads + 255) / 256);
        pool_kernel<<<pblocks, 256, 0, stream>>>(bufB, batch, pooled, cnt, N);
        pool_div_kernel<<<(256 * 128 + 255) / 256, 256, 0, stream>>>(pooled, cnt);
    }

    // head: relu(pooled @ o_w1 + o_b1) @ o_w2 + o_b2
    gemm_node_kernel<<<2, 256, 0, stream>>>(
        pooled, nullptr, o_w1, o_b1, nullptr, nullptr, tbuf, 256, 128, 1);
    gemm_node_kernel<<<2, 256, 0, stream>>>(
        tbuf, nullptr, o_w2, o_b2, nullptr, nullptr, (float*)d_out, 256, 128, 0);
}